// PPSSampler_69870527971642
// MI455X (gfx1250) — compile-verified
//
#include <hip/hip_runtime.h>
#include <stdint.h>

// Problem constants (from reference)
#define BATCH     128
#define CHOICES   50000
#define NSAMP     4
#define ROWS      (BATCH * NSAMP)       // 512
#define KSEL      32
#define TPB       1024                  // 32 waves (wave32)
#define CHUNK_F   4096                  // floats per async chunk (16 KB)
#define NCHUNK    13                    // ceil(50000/4096)
#define ITEMS     (NCHUNK * 4)          // 52 register-resident logits per thread
#define NEG_INF   (-3.402823466e38f)

__global__ __launch_bounds__(TPB) void pps_topk_kernel(
    const float* __restrict__ scores,
    const float* __restrict__ u,
    float* __restrict__ out,
    uint32_t u_total_bytes)
{
    __shared__ float sbuf[2][CHUNK_F];   // 32 KB double buffer for async u staging
    __shared__ float red_v[32];
    __shared__ int   red_j[32];
    __shared__ int   win_j_s;

    const int tid = threadIdx.x;
    const int row = blockIdx.x;                      // 0..511
    const int b   = row & (BATCH - 1);               // row % 128
    const float* __restrict__ srow = scores + (size_t)b * CHOICES;
    float* __restrict__ orow = out + (size_t)row * CHOICES;
    const uint32_t row_u_byte = (uint32_t)row * (uint32_t)(CHOICES * 4);
    const uint32_t glimit = u_total_bytes - 16u;

    // ---- async global -> LDS issue (CDNA5 TDM-style async copy path) ----
    auto issue = [&](int c) {
        uint32_t goff = row_u_byte + (uint32_t)(c * CHUNK_F + 4 * tid) * 4u;
        if (goff > glimit) goff = glimit;            // clamp: stay in-bounds, data unused
        uint32_t loff = (uint32_t)(uintptr_t)(&sbuf[c & 1][4 * tid]); // low 32b of flat = LDS offset
        asm volatile("global_load_async_to_lds_b128 %0, %1, %2"
                     :: "v"(loff), "v"(goff), "s"(u)
                     : "memory");
    };

    issue(0);
    issue(1);

    // ---- zero-fill output row while the first async loads are in flight ----
    {
        float4 z = make_float4(0.f, 0.f, 0.f, 0.f);
        float4* o4 = (float4*)orow;                  // row base is 16B aligned (50000*4 % 16 == 0)
        for (int i = tid; i < CHOICES / 4; i += TPB) o4[i] = z;
    }

    // ---- streamed gumbel + score -> register-resident logits ----
    float vals[ITEMS];
#pragma unroll
    for (int c = 0; c < NCHUNK; ++c) {
        if (c + 1 < NCHUNK) { asm volatile("s_wait_asynccnt 0x1" ::: "memory"); }
        else                { asm volatile("s_wait_asynccnt 0x0" ::: "memory"); }
        __syncthreads();                             // chunk c visible to all waves
#pragma unroll
        for (int q = 0; q < 4; ++q) {
            const int e = c * 4 + q;
            const int j = c * CHUNK_F + q * 1024 + tid;
            float v = NEG_INF;
            if (j < CHOICES) {
                float uu = sbuf[c & 1][q * 1024 + tid];
                float g  = -__logf(-__logf(uu + 1e-10f) + 1e-10f);
                v = srow[j] + g;                     // scores served from L2 (25.6MB << 192MB)
            }
            vals[e] = v;
        }
        __syncthreads();                             // everyone done reading buf[c&1]
        if (c + 2 < NCHUNK) issue(c + 2);            // refill the freed buffer
    }

    // order zero-fill stores (L2) before any 1.0 stores from other waves
    asm volatile("s_wait_storecnt 0x0" ::: "memory");

    // ---- cached per-thread local max (first-index tie-break) ----
    float lv = NEG_INF; int lj = 0;
#pragma unroll
    for (int e = 0; e < ITEMS; ++e) {
        if (vals[e] > lv) { lv = vals[e]; lj = (e >> 2) * CHUNK_F + (e & 3) * 1024 + tid; }
    }

    const int lane = tid & 31;
    const int wv   = tid >> 5;

    // ---- iterative top-32: butterfly wave reduce -> cross-wave reduce -> mask winner ----
    for (int k = 0; k < KSEL; ++k) {
        float v = lv; int j = lj;
#pragma unroll
        for (int off = 16; off > 0; off >>= 1) {
            float ov = __shfl_xor(v, off, 32);
            int   oj = __shfl_xor(j, off, 32);
            if (ov > v || (ov == v && oj < j)) { v = ov; j = oj; }
        }
        if (lane == 0) { red_v[wv] = v; red_j[wv] = j; }
        __syncthreads();
        if (tid < 32) {
            v = red_v[tid]; j = red_j[tid];
#pragma unroll
            for (int off = 16; off > 0; off >>= 1) {
                float ov = __shfl_xor(v, off, 32);
                int   oj = __shfl_xor(j, off, 32);
                if (ov > v || (ov == v && oj < j)) { v = ov; j = oj; }
            }
            if (tid == 0) win_j_s = j;
        }
        __syncthreads();
        const int wj = win_j_s;
        if ((wj & 1023) == tid) {                    // owner lane of the winning element
            orow[wj] = 1.0f;
            const int esel = wj >> 10;               // = c*4 + q
#pragma unroll
            for (int e = 0; e < ITEMS; ++e) if (e == esel) vals[e] = NEG_INF;
            lv = NEG_INF; lj = 0;
#pragma unroll
            for (int e = 0; e < ITEMS; ++e) {
                if (vals[e] > lv) { lv = vals[e]; lj = (e >> 2) * CHUNK_F + (e & 3) * 1024 + tid; }
            }
        }
        // win_j_s safely re-written only after the next iteration's first barrier
    }
}

extern "C" void kernel_launch(void* const* d_in, const int* in_sizes, int n_in,
                              void* d_out, int out_size, void* d_ws, size_t ws_size,
                              hipStream_t stream) {
    const float* scores = (const float*)d_in[0];     // (128, 50000, 1) f32
    const float* u      = (const float*)d_in[1];     // (512, 50000)    f32
    float* out          = (float*)d_out;             // (4, 128, 50000) f32
    const uint32_t u_bytes = (uint32_t)in_sizes[1] * 4u;
    pps_topk_kernel<<<ROWS, TPB, 0, stream>>>(scores, u, out, u_bytes);
}